// GNN_44942537786128
// MI455X (gfx1250) — compile-verified
//
#include <hip/hip_runtime.h>
#include <hip/hip_bf16.h>

// ---------------------------------------------------------------------------
// GCN forward for MI455X (gfx1250, wave32).
//   conv1: h1 = relu( scatter(norm * (x@W1)) + selfloop + b1 )   F: 128 -> 128
//   conv2: h2 =        scatter(norm * (h1@W2)) + selfloop + b2   F: 128 -> 64
//   out  : per-graph mean pool over 64 graphs
// GEMMs use V_WMMA_F32_16X16X4_F32 (full fp32 accuracy, CDNA5 matrix path).
// Aggregation is L2-resident fp32 atomics (outputs fit in 192MB L2).
// ---------------------------------------------------------------------------

typedef float v2f __attribute__((ext_vector_type(2)));
typedef float v8f __attribute__((ext_vector_type(8)));

#ifndef N_GRAPHS_C
#define N_GRAPHS_C 64
#endif

// ---------------- utility kernels ----------------

__global__ void fill_kernel(float* __restrict__ p, float v, size_t n) {
  size_t i = (size_t)blockIdx.x * blockDim.x + threadIdx.x;
  if (i < n) p[i] = v;
}

__global__ void deg_accum_kernel(const int* __restrict__ dst,
                                 float* __restrict__ deg, int E) {
  int e = blockIdx.x * blockDim.x + threadIdx.x;
  if (e < E) unsafeAtomicAdd(&deg[dst[e]], 1.0f);
}

__global__ void rsqrt_kernel(float* __restrict__ d, int N) {
  int i = blockIdx.x * blockDim.x + threadIdx.x;
  if (i < N) d[i] = rsqrtf(d[i]);   // deg >= 1 always (self-loop), no zero guard needed
}

// ---------------- fp32 WMMA GEMM: C[M x (NT*16)] = A[M x K] @ B[K x (NT*16)] ----------------
// One wave computes a 16-row strip across all NT column tiles.
// Fragment layout per CDNA5 ISA 7.12.2:
//   A 16x4 : lane<16 -> K={k0,k0+1}, lane>=16 -> K={k0+2,k0+3}; VGPR0/1 = K lo/hi
//   B 4x16 : mirrored (rows striped across lane halves)
//   C/D    : VGPR r -> row r (lanes 0-15) / row 8+r (lanes 16-31)
template <int NT, int K>
__global__ void gemm_wmma_f32(const float* __restrict__ A,
                              const float* __restrict__ B,
                              float* __restrict__ C, int M) {
  const int N = NT * 16;
  int wave = (blockIdx.x * blockDim.x + threadIdx.x) >> 5;
  int lane = threadIdx.x & 31;
  int m0 = wave * 16;
  if (m0 >= M) return;                 // wave-uniform exit: EXEC stays all-ones

  int l15 = lane & 15;                 // M row (A) / N col (B,C)
  int kp  = lane >> 4;                 // K pair select (0 or 1)

  v8f acc[NT] = {};

  const float* arow = A + (size_t)(m0 + l15) * K;

  for (int k0 = 0; k0 < K; k0 += 4) {
    // A fragment: 2 consecutive K values -> one 8-byte load
    v2f a = *(const v2f*)(arow + k0 + 2 * kp);
#pragma unroll
    for (int n = 0; n < NT; ++n) {
      v2f b;
      b.x = B[(size_t)(k0 + 2 * kp + 0) * N + n * 16 + l15];
      b.y = B[(size_t)(k0 + 2 * kp + 1) * N + n * 16 + l15];
      acc[n] = __builtin_amdgcn_wmma_f32_16x16x4_f32(
          false, a, false, b, (short)0, acc[n], false, false);
    }
  }

  int mh = kp * 8;                     // row-half offset for C/D
#pragma unroll
  for (int n = 0; n < NT; ++n) {
#pragma unroll
    for (int r = 0; r < 8; ++r) {
      C[(size_t)(m0 + mh + r) * N + n * 16 + l15] = acc[n][r];
    }
  }
}

// ---------------- edge scatter: out[dst] += h[src] * dis[src]*dis[dst] ----------------
// F/4 threads per edge, float4 gathers, hardware fp32 atomics (resolve in L2).
template <int F>
__global__ void scatter_edges(const float* __restrict__ h,
                              const float* __restrict__ dis,
                              const int* __restrict__ src,
                              const int* __restrict__ dst,
                              float* __restrict__ out, int E) {
  const int TPE = F / 4;
  long long gid = (long long)blockIdx.x * blockDim.x + threadIdx.x;
  long long e = gid / TPE;
  int t = (int)(gid - e * TPE);
  if (e >= E) return;
  int s = src[e];
  int d = dst[e];
  float norm = dis[s] * dis[d];
  const float4 v = *(const float4*)(h + (size_t)s * F + t * 4);
  float* o = out + (size_t)d * F + t * 4;
  unsafeAtomicAdd(o + 0, v.x * norm);
  unsafeAtomicAdd(o + 1, v.y * norm);
  unsafeAtomicAdd(o + 2, v.z * norm);
  unsafeAtomicAdd(o + 3, v.w * norm);
}

// ---------------- epilogue: add self-loop message + bias, optional relu ----------------
template <int F, bool RELU>
__global__ void finish_conv(const float* __restrict__ h,
                            const float* __restrict__ dis,
                            const float* __restrict__ bias,
                            float* __restrict__ out, int N) {
  long long gid = (long long)blockIdx.x * blockDim.x + threadIdx.x;
  long long n = gid / F;
  int f = (int)(gid - n * F);
  if (n >= N) return;
  float di = dis[n];
  size_t idx = (size_t)n * F + f;
  float v = out[idx] + h[idx] * (di * di) + bias[f];
  out[idx] = RELU ? fmaxf(v, 0.0f) : v;
}

// ---------------- mean pool over graph ids ----------------
__global__ void pool_accum(const float* __restrict__ h,
                           const int* __restrict__ batch,
                           float* __restrict__ sums,
                           float* __restrict__ cnt, int N) {
  long long gid = (long long)blockIdx.x * blockDim.x + threadIdx.x;
  long long n = gid >> 6;              // F_OUT = 64
  int f = (int)(gid & 63);
  if (n >= N) return;
  int g = batch[n];
  unsafeAtomicAdd(&sums[(size_t)g * 64 + f], h[(size_t)n * 64 + f]);
  if (f == 0) unsafeAtomicAdd(&cnt[g], 1.0f);
}

__global__ void pool_final(const float* __restrict__ sums,
                           const float* __restrict__ cnt,
                           float* __restrict__ out) {
  int gid = blockIdx.x * blockDim.x + threadIdx.x;   // 64*64 threads
  if (gid >= N_GRAPHS_C * 64) return;
  int g = gid >> 6;
  out[gid] = sums[gid] / fmaxf(cnt[g], 1.0f);
}

// ---------------- host launcher ----------------

extern "C" void kernel_launch(void* const* d_in, const int* in_sizes, int n_in,
                              void* d_out, int out_size, void* d_ws, size_t ws_size,
                              hipStream_t stream) {
  const float* x     = (const float*)d_in[0];
  const int*   eidx  = (const int*)d_in[1];    // [2, E]
  const int*   batch = (const int*)d_in[2];    // [N]
  const float* W1    = (const float*)d_in[3];
  const float* b1    = (const float*)d_in[4];
  const float* W2    = (const float*)d_in[5];
  const float* b2    = (const float*)d_in[6];

  const int F1 = 128, F2 = 64;
  const int N = in_sizes[0] / F1;        // 100000
  const int E = in_sizes[1] / 2;         // 1600000
  const int* src = eidx;
  const int* dst = eidx + E;

  // Workspace carving (floats)
  float* ws   = (float*)d_ws;
  float* dis  = ws;                                  // N       (deg -> deg^-1/2)
  float* h    = dis + (((size_t)N + 255) & ~(size_t)255); // N*128 (h1, then reused for h2)
  float* agg1 = h    + (size_t)N * F1;               // N*128
  float* agg2 = agg1 + (size_t)N * F1;               // N*64
  float* sums = agg2 + (size_t)N * F2;               // 64*64
  float* cnt  = sums + (size_t)N_GRAPHS_C * F2;      // 64
  (void)ws_size; (void)n_in; (void)out_size;

  const int TB = 256;
  auto blocks = [](long long total, int tb) { return (unsigned)((total + tb - 1) / tb); };

  // 1) degree (init 1.0 for self-loop) -> dis = deg^-1/2
  fill_kernel<<<blocks(N, TB), TB, 0, stream>>>(dis, 1.0f, (size_t)N);
  deg_accum_kernel<<<blocks(E, TB), TB, 0, stream>>>(dst, dis, E);
  rsqrt_kernel<<<blocks(N, TB), TB, 0, stream>>>(dis, N);

  // 2) h1 = x @ W1   (M=100000, K=128, N=128 -> 8 column tiles)
  {
    int mt = (N + 15) / 16;                // 16-row strips, one per wave
    gemm_wmma_f32<8, 128><<<blocks((long long)mt * 32, TB), TB, 0, stream>>>(x, W1, h, N);
  }

  // 3) agg1 = scatter(h1) ; += selfloop + b1 ; relu
  fill_kernel<<<blocks((long long)N * F1, TB), TB, 0, stream>>>(agg1, 0.0f, (size_t)N * F1);
  scatter_edges<128><<<blocks((long long)E * (F1 / 4), TB), TB, 0, stream>>>(h, dis, src, dst, agg1, E);
  finish_conv<128, true><<<blocks((long long)N * F1, TB), TB, 0, stream>>>(h, dis, b1, agg1, N);

  // 4) h2 = agg1 @ W2   (K=128, N=64 -> 4 column tiles); reuse h buffer
  {
    int mt = (N + 15) / 16;
    gemm_wmma_f32<4, 128><<<blocks((long long)mt * 32, TB), TB, 0, stream>>>(agg1, W2, h, N);
  }

  // 5) agg2 = scatter(h2) ; += selfloop + b2 (no relu)
  fill_kernel<<<blocks((long long)N * F2, TB), TB, 0, stream>>>(agg2, 0.0f, (size_t)N * F2);
  scatter_edges<64><<<blocks((long long)E * (F2 / 4), TB), TB, 0, stream>>>(h, dis, src, dst, agg2, E);
  finish_conv<64, false><<<blocks((long long)N * F2, TB), TB, 0, stream>>>(h, dis, b2, agg2, N);

  // 6) mean pool
  fill_kernel<<<blocks(N_GRAPHS_C * F2 + N_GRAPHS_C, TB), TB, 0, stream>>>(
      sums, 0.0f, (size_t)(N_GRAPHS_C * F2 + N_GRAPHS_C));
  pool_accum<<<blocks((long long)N * F2, TB), TB, 0, stream>>>(agg2, batch, sums, cnt, N);
  pool_final<<<blocks(N_GRAPHS_C * F2, TB), TB, 0, stream>>>(sums, cnt, (float*)d_out);
}